// GCN_res_46780783788518
// MI455X (gfx1250) — compile-verified
//
#include <hip/hip_runtime.h>
#include <hip/hip_bf16.h>
#include <math.h>

typedef __attribute__((ext_vector_type(2))) float v2f;
typedef __attribute__((ext_vector_type(8))) float v8f;

#define GN 100000
#define GE 300000
#define GH 256
#define GL 6

__device__ __forceinline__ void atomic_add_f32(float* p, float v) {
    __hip_atomic_fetch_add(p, v, __ATOMIC_RELAXED, __HIP_MEMORY_SCOPE_AGENT);
}

// ---------------------------------------------------------------------------
// Degree / symmetric norm
// ---------------------------------------------------------------------------
__global__ void deg_init(float* deg, int n) {
    int i = blockIdx.x * 256 + threadIdx.x;
    if (i < n) deg[i] = 1.0f;               // self loop
}
__global__ void deg_edges(const int* __restrict__ ei, float* deg, int e) {
    int i = blockIdx.x * 256 + threadIdx.x;
    if (i < e) atomic_add_f32(&deg[ei[e + i]], 1.0f);   // dst = ei[1][i]
}
__global__ void deg_finish(float* deg, int n) {
    int i = blockIdx.x * 256 + threadIdx.x;
    if (i < n) deg[i] = rsqrtf(fmaxf(deg[i], 1.0f));
}

// softmax over the L residual-mix weights (tiny, single thread)
__global__ void softmax_res(const float* __restrict__ rw, float* wl) {
    float mx = rw[0];
    for (int i = 1; i < GL; ++i) mx = fmaxf(mx, rw[i]);
    float s = 0.f;
    float e[GL];
    for (int i = 0; i < GL; ++i) { e[i] = __expf(rw[i] - mx); s += e[i]; }
    for (int i = 0; i < GL; ++i) wl[i] = e[i] / s;
}

// ---------------------------------------------------------------------------
// WMMA fp32 GEMM: C[nrows,256] = A[nrows,K] @ B[K,256] (+bias), optional copy C2
// Block: 256 threads = 8 waves; wave = 16 rows x 64 cols; block = 32 rows x 256 cols
// Boundary rows: clamp pointer (stay in-bounds) and multiply fragment by 0/1
// mask -> unconditional loads, no exec-mask churn in the inner loop.
// ---------------------------------------------------------------------------
template <int K>
__global__ __launch_bounds__(256) void gemm_wmma(const float* __restrict__ A,
                                                 const float* __restrict__ B,
                                                 const float* __restrict__ bias,
                                                 float* __restrict__ C,
                                                 float* __restrict__ C2,
                                                 int nrows) {
    const int NCOL = 256;
    const int lane = threadIdx.x & 31;
    const int wave = threadIdx.x >> 5;
    const int waveRow = wave >> 2;          // 0..1
    const int waveCol = wave & 3;           // 0..3
    const int rbase = blockIdx.x * 32 + waveRow * 16;
    const int cbase = waveCol * 64;
    const int m  = lane & 15;               // row within A-tile / col within B-tile
    const int kh = lane >> 4;               // K-half select (0 or 1)

    const int arow = rbase + m;
    const float amask = (arow < nrows) ? 1.0f : 0.0f;
    const float* Arow = A + (size_t)min(arow, nrows - 1) * K;

    v8f acc[4] = {};

    for (int k = 0; k < K; k += 4) {
        const int ka = k + kh * 2;
        v2f a = *(const v2f*)(Arow + ka);   // A[m][ka], A[m][ka+1] (clamped row)
        a.x *= amask;
        a.y *= amask;
        const float* B0 = B + (size_t)ka * NCOL + cbase + m;
        const float* B1 = B0 + NCOL;
#pragma unroll
        for (int t = 0; t < 4; ++t) {
            v2f b;
            b.x = B0[t * 16];               // B[ka  ][col]
            b.y = B1[t * 16];               // B[ka+1][col]
            acc[t] = __builtin_amdgcn_wmma_f32_16x16x4_f32(
                false, a, false, b, (short)0, acc[t], false, false);
        }
    }

#pragma unroll
    for (int t = 0; t < 4; ++t) {
        const int col = cbase + t * 16 + m;
        const float bv = bias ? bias[col] : 0.0f;
#pragma unroll
        for (int v = 0; v < 8; ++v) {
            const int row = rbase + v + kh * 8;
            if (row < nrows) {
                const float val = acc[t][v] + bv;
                C[(size_t)row * NCOL + col] = val;
                if (C2) C2[(size_t)row * NCOL + col] = val;
            }
        }
    }
}

// ---------------------------------------------------------------------------
// Edge scatter: agg[dst] += hw[src] * dinv[src]*dinv[dst]   (float4 per thread)
// ---------------------------------------------------------------------------
__global__ __launch_bounds__(256) void scatter_edges(const int* __restrict__ ei,
                                                     const float* __restrict__ hw,
                                                     const float* __restrict__ dinv,
                                                     float* __restrict__ agg, int e) {
    int t = blockIdx.x * 256 + threadIdx.x;     // E*64 threads
    int ed = t >> 6;
    if (ed >= e) return;
    int f = (t & 63) << 2;
    int src = ei[ed];
    int dst = ei[e + ed];
    float nrm = dinv[src] * dinv[dst];
    const float4 hv = *(const float4*)(hw + (size_t)src * GH + f);
    float* ap = agg + (size_t)dst * GH + f;
    atomic_add_f32(ap + 0, hv.x * nrm);
    atomic_add_f32(ap + 1, hv.y * nrm);
    atomic_add_f32(ap + 2, hv.z * nrm);
    atomic_add_f32(ap + 3, hv.w * nrm);
}

// ---------------------------------------------------------------------------
// BatchNorm stats: thread = feature column, coalesced row-major sweep
// value = agg + hw*dinv^2 (self loop) + conv_b
// ---------------------------------------------------------------------------
__global__ __launch_bounds__(256) void bn_stats(const float* __restrict__ agg,
                                                const float* __restrict__ hw,
                                                const float* __restrict__ dinv,
                                                const float* __restrict__ cb,
                                                float* __restrict__ sums,
                                                float* __restrict__ sumsq,
                                                int n, int rpb) {
    int f = threadIdx.x;
    int r0 = blockIdx.x * rpb;
    int r1 = min(r0 + rpb, n);
    float b = cb[f];
    float s = 0.f, q = 0.f;
    for (int r = r0; r < r1; ++r) {
        float d = dinv[r];
        size_t i = (size_t)r * GH + f;
        float v = agg[i] + hw[i] * d * d + b;
        s += v;
        q += v * v;
    }
    atomic_add_f32(&sums[f], s);
    atomic_add_f32(&sumsq[f], q);
}

__global__ void bn_finalize(const float* __restrict__ sums,
                            const float* __restrict__ sumsq,
                            const float* __restrict__ gamma,
                            const float* __restrict__ beta,
                            float* __restrict__ scale,
                            float* __restrict__ shift, int n) {
    int f = threadIdx.x;
    float inv_n = 1.0f / (float)n;
    float mean = sums[f] * inv_n;
    float var = sumsq[f] * inv_n - mean * mean;
    float is = rsqrtf(var + 1e-5f);
    float sc = gamma[f] * is;
    scale[f] = sc;
    shift[f] = beta[f] - mean * sc;
}

// ---------------------------------------------------------------------------
// Fused BN-apply + ReLU + residual + weighted mix accumulation.
// Writes new h IN PLACE over agg.
// ---------------------------------------------------------------------------
__global__ __launch_bounds__(256) void apply_layer(float* __restrict__ agg,
                                                   const float* __restrict__ hw,
                                                   const float* __restrict__ dinv,
                                                   const float* __restrict__ cb,
                                                   const float* __restrict__ scale,
                                                   const float* __restrict__ shift,
                                                   const float* __restrict__ xin,
                                                   const float* __restrict__ hprev,
                                                   float* __restrict__ mix,
                                                   const float* __restrict__ wlp,
                                                   int n) {
    int idx = blockIdx.x * 256 + threadIdx.x;   // n*256 threads exactly
    int node = idx >> 8;
    int f = idx & 255;
    if (node >= n) return;
    float d = dinv[node];
    float v = agg[idx] + hw[idx] * d * d + cb[f];
    float a = fmaxf(v * scale[f] + shift[f], 0.0f);
    float hn = a + 0.2f * xin[idx];
    if (hprev) hn += 0.7f * hprev[idx];
    agg[idx] = hn;                               // becomes next layer's h
    mix[idx] += wlp[0] * hn;
}

// ---------------------------------------------------------------------------
// Output: logits = mix @ out_w + out_b, then row-wise log-softmax.
// One wave per node (wave32), out_w staged in LDS with pad-41 stride.
// ---------------------------------------------------------------------------
__global__ __launch_bounds__(256) void out_logsoftmax(const float* __restrict__ mix,
                                                      const float* __restrict__ W,
                                                      const float* __restrict__ bias,
                                                      float* __restrict__ out,
                                                      int nrows) {
    __shared__ float lw[256 * 41];
    for (int i = threadIdx.x; i < 256 * 40; i += 256) {
        int k = i / 40, c = i % 40;
        lw[k * 41 + c] = W[i];
    }
    __syncthreads();

    const int lane = threadIdx.x & 31;
    const int wave = threadIdx.x >> 5;
    const int node = blockIdx.x * 8 + wave;
    if (node >= nrows) return;

    float acc[40];
#pragma unroll
    for (int c = 0; c < 40; ++c) acc[c] = 0.f;

    const float* xr = mix + (size_t)node * GH;
#pragma unroll
    for (int j = 0; j < 8; ++j) {
        int k = j * 32 + lane;
        float xv = xr[k];
        const float* wr = &lw[k * 41];
#pragma unroll
        for (int c = 0; c < 40; ++c) acc[c] += xv * wr[c];
    }
    // butterfly reduce across the wave (all lanes end with the full sums)
#pragma unroll
    for (int c = 0; c < 40; ++c) {
#pragma unroll
        for (int s = 16; s > 0; s >>= 1) acc[c] += __shfl_xor(acc[c], s, 32);
    }
    float mx = -3.4e38f;
#pragma unroll
    for (int c = 0; c < 40; ++c) { acc[c] += bias[c]; mx = fmaxf(mx, acc[c]); }
    float s = 0.f;
#pragma unroll
    for (int c = 0; c < 40; ++c) s += __expf(acc[c] - mx);
    float lse = mx + __logf(s);
    for (int c = lane; c < 40; c += 32)
        out[(size_t)node * 40 + c] = acc[c] - lse;
}

// ---------------------------------------------------------------------------
extern "C" void kernel_launch(void* const* d_in, const int* in_sizes, int n_in,
                              void* d_out, int out_size, void* d_ws, size_t ws_size,
                              hipStream_t stream) {
    const float* x      = (const float*)d_in[0];
    const int*   ei     = (const int*)  d_in[1];
    const float* w_in   = (const float*)d_in[2];
    const float* b_in   = (const float*)d_in[3];
    const float* conv_w = (const float*)d_in[4];
    const float* conv_b = (const float*)d_in[5];
    const float* gamma  = (const float*)d_in[6];
    const float* beta   = (const float*)d_in[7];
    const float* out_w  = (const float*)d_in[8];
    const float* out_b  = (const float*)d_in[9];
    const float* res_w  = (const float*)d_in[10];
    float* out = (float*)d_out;

    const int N = GN, E = GE, H = GH, L = GL;
    const size_t NH = (size_t)N * H;

    char* w = (char*)d_ws;
    float* bufA = (float*)(w);                 // h ping
    float* bufB = (float*)(w + 1 * NH * 4);    // h pong / agg
    float* hw   = (float*)(w + 2 * NH * 4);
    float* xin  = (float*)(w + 3 * NH * 4);
    float* mix  = (float*)(w + 4 * NH * 4);
    float* dinv = (float*)(w + 5 * NH * 4);
    float* sums  = dinv + N;                   // 256
    float* sumsq = sums + 256;                 // 256 (zeroed with sums)
    float* scale = sumsq + 256;
    float* shift = scale + 256;
    float* wl    = shift + 256;                // L softmax weights

    hipMemsetAsync(mix, 0, NH * 4, stream);

    deg_init  <<<(N + 255) / 256, 256, 0, stream>>>(dinv, N);
    deg_edges <<<(E + 255) / 256, 256, 0, stream>>>(ei, dinv, E);
    deg_finish<<<(N + 255) / 256, 256, 0, stream>>>(dinv, N);
    softmax_res<<<1, 1, 0, stream>>>(res_w, wl);

    const int gemm_blocks = (N + 31) / 32;     // 3125
    // input FC: h0 = x @ w_in + b_in  (also keep a copy as x_input)
    gemm_wmma<128><<<gemm_blocks, 256, 0, stream>>>(x, w_in, b_in, bufA, xin, N);

    float* hcur = bufA;
    float* agg  = bufB;
    const int rpb = (N + 255) / 256;           // rows per bn_stats block

    for (int l = 0; l < L; ++l) {
        gemm_wmma<256><<<gemm_blocks, 256, 0, stream>>>(
            hcur, conv_w + (size_t)l * H * H, nullptr, hw, nullptr, N);

        hipMemsetAsync(agg, 0, NH * 4, stream);
        hipMemsetAsync(sums, 0, 2 * 256 * 4, stream);

        scatter_edges<<<(E * 64) / 256, 256, 0, stream>>>(ei, hw, dinv, agg, E);

        bn_stats<<<256, 256, 0, stream>>>(agg, hw, dinv, conv_b + l * H,
                                          sums, sumsq, N, rpb);
        bn_finalize<<<1, 256, 0, stream>>>(sums, sumsq, gamma + l * H,
                                           beta + l * H, scale, shift, N);

        apply_layer<<<(int)(NH / 256), 256, 0, stream>>>(
            agg, hw, dinv, conv_b + l * H, scale, shift, xin,
            (l > 0 ? hcur : nullptr), mix, wl + l, N);

        float* t = hcur; hcur = agg; agg = t;  // new h lives in old agg buffer
    }

    out_logsoftmax<<<(N + 7) / 8, 256, 0, stream>>>(mix, out_w, out_b, out, N);
}